// AttnDecoderRNN_15539191677010
// MI455X (gfx1250) — compile-verified
//
#include <hip/hip_runtime.h>
#include <hip/hip_bf16.h>
#include <math.h>

// ---------------------------------------------------------------------------
// AttnDecoderRNN single step for MI455X (gfx1250, wave32).
// Memory-bound GEMV workload (~241 MB of fp32 weights streamed once
// -> ~10.4us floor at 23.3 TB/s). Matrix path: V_WMMA_F32_16X16X4_F32.
// ---------------------------------------------------------------------------

#define HIDDEN 1024
#define EMBED  1024
#define VOCAB  50257
#define MAXLEN 100
#define ECAT   (EMBED + HIDDEN)   // 2048

typedef __attribute__((ext_vector_type(2))) float v2f;
typedef __attribute__((ext_vector_type(8))) float v8f;

// ---------------------------------------------------------------------------
// WMMA GEMV: each wave computes 16 rows of y = W @ x + b; 4 waves per block
// (64 rows/block), x staged once per block into LDS.
//
// Operand layout for V_WMMA_F32_16X16X4_F32 (ISA 7.12.2):
//   A (16x4, 2 VGPRs): lanes 0-15 = rows, k-pair (k0,k0+1);
//                      lanes 16-31 = rows, k-pair (k0+2,k0+3).
//   B (4x16): only column N=0 is live -> lanes 0 and 16 carry x, rest 0.
//     Branch-free: every lane loads the same LDS chunk at the A k-offset and
//     multiplies by mask (1.0 iff lane==0 or lane==16). No EXEC manipulation,
//     so the loop stays uniform and the WMMA EXEC==all-ones rule holds.
//   D column N=0: lane 0 holds M=0..7 in c[0..7], lane 16 holds M=8..15.
// Tail rows: clamp the *load* row, guard only the store.
// ---------------------------------------------------------------------------
__launch_bounds__(128)
__global__ void gemv16_wmma_kernel(const float* __restrict__ W,
                                   const float* __restrict__ bias,
                                   const float* __restrict__ x,
                                   float* __restrict__ y,
                                   int M, int K, int act /*0=none,1=relu*/) {
  __shared__ float xs[ECAT];
  const int tid = threadIdx.x;
  for (int i = tid; i < K; i += 128) xs[i] = x[i];
  __syncthreads();

  const int wave = tid >> 5, lane = tid & 31;
  const int row0 = blockIdx.x * 64 + wave * 16;

  int r = row0 + (lane & 15);
  if (r > M - 1) r = M - 1;               // clamp: EXEC stays all-ones
  const int koff = (lane >> 4) << 1;      // 0 for lanes 0-15, 2 for 16-31
  const float* aptr = W + (size_t)r * K + koff;
  const float* bptr = xs + koff;
  const float msk = ((lane & 15) == 0) ? 1.0f : 0.0f;  // lanes 0,16 -> N=0

  v8f c = {};
  for (int k0 = 0; k0 < K; k0 += 16) {    // 4 WMMA steps per trip
    // streaming weights: non-temporal (out_W is 206MB > 192MB L2)
    v2f a0 = __builtin_nontemporal_load((const v2f*)(aptr + k0));
    v2f a1 = __builtin_nontemporal_load((const v2f*)(aptr + k0 + 4));
    v2f a2 = __builtin_nontemporal_load((const v2f*)(aptr + k0 + 8));
    v2f a3 = __builtin_nontemporal_load((const v2f*)(aptr + k0 + 12));
    v2f b0 = *(const v2f*)(bptr + k0)      * msk;   // LDS broadcast + mask
    v2f b1 = *(const v2f*)(bptr + k0 + 4)  * msk;
    v2f b2 = *(const v2f*)(bptr + k0 + 8)  * msk;
    v2f b3 = *(const v2f*)(bptr + k0 + 12) * msk;
    __builtin_prefetch(aptr + k0 + 128, 0, 1);      // global_prefetch_b8
    c = __builtin_amdgcn_wmma_f32_16x16x4_f32(false, a0, false, b0, (short)0, c, false, false);
    c = __builtin_amdgcn_wmma_f32_16x16x4_f32(false, a1, false, b1, (short)0, c, false, false);
    c = __builtin_amdgcn_wmma_f32_16x16x4_f32(false, a2, false, b2, (short)0, c, false, false);
    c = __builtin_amdgcn_wmma_f32_16x16x4_f32(false, a3, false, b3, (short)0, c, false, false);
  }

  if ((lane & 15) == 0) {                 // lanes 0 and 16 own column N=0
    int mbase = row0 + ((lane >> 4) << 3);
    #pragma unroll
    for (int i = 0; i < 8; ++i) {
      int m = mbase + i;
      if (m < M) {
        float v = c[i] + bias[m];
        if (act == 1) v = fmaxf(v, 0.0f);
        y[m] = v;
      }
    }
  }
}

// ---------------------------------------------------------------------------
// Attention front end (single block, 1024 threads = 32 waves):
// embedding gather, 100x2048 attention GEMV, softmax(100), 100x1024 apply.
// ---------------------------------------------------------------------------
__launch_bounds__(1024)
__global__ void attn_kernel(const int* __restrict__ tok,
                            const float* __restrict__ hidden,
                            const float* __restrict__ enc,
                            const float* __restrict__ emb,
                            const float* __restrict__ attn_W,
                            const float* __restrict__ attn_b,
                            float* __restrict__ x_cat,      // [2048]
                            float* __restrict__ attn_out) { // [100]
  __shared__ float emb_s[EMBED];
  __shared__ float hid_s[HIDDEN];
  __shared__ float w_s[MAXLEN];
  const int tid = threadIdx.x;
  const int t = tok[0];                   // token (int64 low word on LE)

  float e = emb[(size_t)t * EMBED + tid];
  emb_s[tid] = e;
  x_cat[tid] = e;                         // cat([embedded, attn_applied]) lhs
  hid_s[tid] = hidden[tid];
  __syncthreads();

  // attn logits: wave w handles rows w, w+32, w+64, w+96 (<100)
  const int wave = tid >> 5, lane = tid & 31;
  for (int row = wave; row < MAXLEN; row += 32) {
    const float* wr = attn_W + (size_t)row * ECAT;
    float acc = 0.0f;
    for (int k = lane; k < EMBED; k += 32)
      acc += wr[k] * emb_s[k] + wr[EMBED + k] * hid_s[k];
    for (int off = 16; off; off >>= 1) acc += __shfl_xor(acc, off, 32);
    if (lane == 0) w_s[row] = acc + attn_b[row];
  }
  __syncthreads();

  if (tid == 0) {                         // softmax over 100 (tiny, serial)
    float mx = w_s[0];
    for (int i = 1; i < MAXLEN; ++i) mx = fmaxf(mx, w_s[i]);
    float s = 0.0f;
    for (int i = 0; i < MAXLEN; ++i) { float ev = __expf(w_s[i] - mx); w_s[i] = ev; s += ev; }
    float inv = 1.0f / s;
    for (int i = 0; i < MAXLEN; ++i) w_s[i] *= inv;
  }
  __syncthreads();

  if (tid < MAXLEN) attn_out[tid] = w_s[tid];

  // attn_applied[j] = sum_l w[l] * enc[l][j]
  float acc = 0.0f;
  for (int l = 0; l < MAXLEN; ++l) acc += w_s[l] * enc[l * HIDDEN + tid];
  x_cat[EMBED + tid] = acc;
}

// ---------------------------------------------------------------------------
// GRU gate combine (PyTorch order r,z,n). gi/gh already include biases.
// ---------------------------------------------------------------------------
__global__ void gru_kernel(const float* __restrict__ gi,
                           const float* __restrict__ gh,
                           const float* __restrict__ h,
                           float* __restrict__ h_out,
                           float* __restrict__ h_out2) {
  int i = blockIdx.x * blockDim.x + threadIdx.x;
  if (i >= HIDDEN) return;
  float r = 1.0f / (1.0f + __expf(-(gi[i]            + gh[i])));
  float z = 1.0f / (1.0f + __expf(-(gi[HIDDEN + i]   + gh[HIDDEN + i])));
  float n = tanhf(gi[2 * HIDDEN + i] + r * gh[2 * HIDDEN + i]);
  float hn = (1.0f - z) * n + z * h[i];
  h_out[i]  = hn;
  h_out2[i] = hn;
}

// ---------------------------------------------------------------------------
// In-place log_softmax over N=50257 (single block, 3 streaming passes over ~201KB).
// ---------------------------------------------------------------------------
__launch_bounds__(1024)
__global__ void logsoftmax_kernel(float* __restrict__ logits, int N) {
  __shared__ float red[32];
  const int tid = threadIdx.x, wave = tid >> 5, lane = tid & 31;

  float mx = -INFINITY;
  for (int i = tid; i < N; i += 1024) mx = fmaxf(mx, logits[i]);
  for (int off = 16; off; off >>= 1) mx = fmaxf(mx, __shfl_xor(mx, off, 32));
  if (lane == 0) red[wave] = mx;
  __syncthreads();
  {
    float v = red[lane];
    for (int off = 16; off; off >>= 1) v = fmaxf(v, __shfl_xor(v, off, 32));
    mx = v;
  }
  __syncthreads();

  float s = 0.0f;
  for (int i = tid; i < N; i += 1024) s += __expf(logits[i] - mx);
  for (int off = 16; off; off >>= 1) s += __shfl_xor(s, off, 32);
  if (lane == 0) red[wave] = s;
  __syncthreads();
  {
    float v = red[lane];
    for (int off = 16; off; off >>= 1) v += __shfl_xor(v, off, 32);
    s = v;
  }
  const float lz = mx + __logf(s);
  for (int i = tid; i < N; i += 1024) logits[i] -= lz;
}

// ---------------------------------------------------------------------------
// Host launcher.  Inputs (setup_inputs order):
//  0 input_token 1 hidden 2 encoder_outputs 3 emb 4 attn_W 5 attn_b
//  6 comb_W 7 comb_b 8 W_ih 9 W_hh 10 b_ih 11 b_hh 12 out_W 13 out_b
// Output: [logp 50257 | h_new 1024 | attn_weights 100] = 51381 fp32.
// ---------------------------------------------------------------------------
extern "C" void kernel_launch(void* const* d_in, const int* in_sizes, int n_in,
                              void* d_out, int out_size, void* d_ws, size_t ws_size,
                              hipStream_t stream) {
  const int*   tok     = (const int*)  d_in[0];
  const float* hidden  = (const float*)d_in[1];
  const float* enc     = (const float*)d_in[2];
  const float* emb     = (const float*)d_in[3];
  const float* attn_W  = (const float*)d_in[4];
  const float* attn_b  = (const float*)d_in[5];
  const float* comb_W  = (const float*)d_in[6];
  const float* comb_b  = (const float*)d_in[7];
  const float* W_ih    = (const float*)d_in[8];
  const float* W_hh    = (const float*)d_in[9];
  const float* b_ih    = (const float*)d_in[10];
  const float* b_hh    = (const float*)d_in[11];
  const float* out_W   = (const float*)d_in[12];
  const float* out_b   = (const float*)d_in[13];

  float* out   = (float*)d_out;
  float* logp  = out;                  // [50257]
  float* h_new = out + VOCAB;          // [1024]
  float* attnw = out + VOCAB + HIDDEN; // [100]

  // workspace (floats): x_cat[2048] | x[1024] | gi[3072] | gh[3072] | hnew[1024]
  float* ws    = (float*)d_ws;
  float* x_cat = ws;
  float* xv    = ws + ECAT;
  float* gi    = xv + HIDDEN;
  float* gh    = gi + 3 * HIDDEN;
  float* hnew  = gh + 3 * HIDDEN;

  // 1) embedding + attention + softmax + apply
  attn_kernel<<<1, 1024, 0, stream>>>(tok, hidden, enc, emb, attn_W, attn_b,
                                      x_cat, attnw);
  // 2) x = relu(comb_W @ x_cat + comb_b)           [1024 x 2048]
  gemv16_wmma_kernel<<<HIDDEN / 64, 128, 0, stream>>>(comb_W, comb_b, x_cat, xv,
                                                      HIDDEN, ECAT, 1);
  // 3) gi = W_ih @ x + b_ih                        [3072 x 1024]
  gemv16_wmma_kernel<<<(3 * HIDDEN) / 64, 128, 0, stream>>>(W_ih, b_ih, xv, gi,
                                                            3 * HIDDEN, HIDDEN, 0);
  // 4) gh = W_hh @ h + b_hh                        [3072 x 1024]
  gemv16_wmma_kernel<<<(3 * HIDDEN) / 64, 128, 0, stream>>>(W_hh, b_hh, hidden, gh,
                                                            3 * HIDDEN, HIDDEN, 0);
  // 5) GRU combine -> h_new (to d_out and ws)
  gru_kernel<<<(HIDDEN + 255) / 256, 256, 0, stream>>>(gi, gh, hidden, h_new, hnew);
  // 6) logits = out_W @ h_new + out_b              [50257 x 1024] — dominant
  gemv16_wmma_kernel<<<(VOCAB + 63) / 64, 128, 0, stream>>>(out_W, out_b, hnew, logp,
                                                            VOCAB, HIDDEN, 0);
  // 7) in-place log_softmax over 50257
  logsoftmax_kernel<<<1, 1024, 0, stream>>>(logp, VOCAB);
}